// Model_60739427501066
// MI455X (gfx1250) — compile-verified
//
#include <hip/hip_runtime.h>

typedef float v2f __attribute__((ext_vector_type(2)));
typedef float v8f __attribute__((ext_vector_type(8)));

#define GROUPS   8
#define MID      64               // K dimension
#define KPAIRS   (MID / 2)        // 32 interleaved K-pair rows
#define BLOCK_M  128
#define BLOCK_N  128
// Dwords per K-pair row: 2*BLOCK_N data + 32 pad so the two half-wave K-pair
// rows (half=0 vs half=1) land in disjoint LDS bank halves (288 % 64 == 32).
#define PAIR_STRIDE (2 * BLOCK_N + 32)

// D = dequant(W4) @ Act, M=N0, K=64, N=N1.
// One block: 8 waves, 128M x 128N tile. Wave w owns rows [w*16, w*16+16).
// LDS layout: element Act[k][n] at Bs[(k>>1)*PAIR_STRIDE + 2n + (k&1)]
// so each WMMA B fragment {B[k0][n], B[k0+1][n]} is ONE aligned ds_load_b64.
__global__ __launch_bounds__(256) void w4a32_wmma_gemm(
    const float* __restrict__ scale,    // [N0, 8]
    const int*   __restrict__ offset,   // [N0]   8 packed 4-bit offsets
    const int*   __restrict__ weight,   // [N0, 8] each word = 8 nibbles
    const float* __restrict__ act,      // [64, N1]
    float* __restrict__ out,            // [N0, N1]
    int N1)
{
    __shared__ float Bs[KPAIRS * PAIR_STRIDE];

    const int tid     = threadIdx.x;
    const int lane    = tid & 31;
    const int wave    = tid >> 5;        // 0..7
    const int m_local = lane & 15;       // M row / N col within 16x16 tile
    const int half    = lane >> 4;       // 0: K {k0,k0+1}; 1: K {k0+2,k0+3}

    const int blockM = blockIdx.y * BLOCK_M;
    const int blockN = blockIdx.x * BLOCK_N;

    // ---- Stage activation tile [64 x 128] into K-pair-interleaved LDS ----
    // 32 kpair-rows * 32 float4-cols = 1024 chunks across 256 threads.
    for (int i = tid; i < KPAIRS * (BLOCK_N / 4); i += 256) {
        const int r  = i >> 5;       // kpair row (K = 2r, 2r+1)
        const int c4 = i & 31;       // float4 column group
        const float* g0 = act + (size_t)(2 * r    ) * N1 + blockN + c4 * 4;
        const float* g1 = act + (size_t)(2 * r + 1) * N1 + blockN + c4 * 4;
        const float4 va = *(const float4*)g0;
        const float4 vb = *(const float4*)g1;
        float4 lo, hi;
        lo.x = va.x; lo.y = vb.x; lo.z = va.y; lo.w = vb.y;
        hi.x = va.z; hi.y = vb.z; hi.z = va.w; hi.w = vb.w;
        float* dst = &Bs[r * PAIR_STRIDE + c4 * 8];
        *(float4*)(dst)     = lo;
        *(float4*)(dst + 4) = hi;
    }

    // ---- Dequantize this wave's 16 rows into 16 A-fragments (v2f each) ----
    // A 16x4 f32 layout: lane m holds A[m][k0] in .x, A[m][k0+1] in .y,
    // k0 = 4*j + 2*half for k-step j.
    const int row = blockM + wave * 16 + m_local;
    const int*   wrow = weight + (size_t)row * GROUPS;
    const float* srow = scale  + (size_t)row * GROUPS;
    const int    offw = offset[row];

    v2f afrag[16];
#pragma unroll
    for (int j = 0; j < 16; ++j) {
        const int k0 = 4 * j + 2 * half;     // even; k0, k0+1 share a group
        const int g  = k0 >> 3;
        const int v0 = k0 & 7;
        const float s  = srow[g];
        const float of = (float)((offw >> (4 * g)) & 15);
        const int   w  = wrow[g];
        const float w0 = (float)((w >> (4 * v0)) & 15);
        const float w1 = (float)((w >> (4 * (v0 + 1))) & 15);
        afrag[j].x = s * (w0 - of);
        afrag[j].y = s * (w1 - of);
    }

    __syncthreads();

    // ---- 16 k-steps x 8 N-subtiles of V_WMMA_F32_16X16X4_F32 ----
    v8f acc[8];
#pragma unroll
    for (int nb = 0; nb < 8; ++nb)
        acc[nb] = (v8f){0.f, 0.f, 0.f, 0.f, 0.f, 0.f, 0.f, 0.f};

#pragma unroll
    for (int j = 0; j < 16; ++j) {
        const v2f a = afrag[j];
        // kpair row for this lane: 2j + half  ->  K = {4j+2*half, 4j+2*half+1}
        const float* brow = Bs + (2 * j + half) * PAIR_STRIDE + 2 * m_local;
#pragma unroll
        for (int nb = 0; nb < 8; ++nb) {
            const v2f b = *(const v2f*)(brow + nb * 32);   // one ds_load_b64
            // (neg_a, A, neg_b, B, c_mod, C, reuse_a, reuse_b)
            acc[nb] = __builtin_amdgcn_wmma_f32_16x16x4_f32(
                false, a, false, b, (short)0, acc[nb], false, false);
        }
    }

    // ---- Store: C/D layout -> VGPR r is row (r + 8*half), col = nb*16+m_local
    float* obase = out + (size_t)(blockM + wave * 16 + half * 8) * N1
                       + blockN + m_local;
#pragma unroll
    for (int nb = 0; nb < 8; ++nb) {
#pragma unroll
        for (int r = 0; r < 8; ++r) {
            obase[(size_t)r * N1 + nb * 16] = acc[nb][r];
        }
    }
}

extern "C" void kernel_launch(void* const* d_in, const int* in_sizes, int n_in,
                              void* d_out, int out_size, void* d_ws, size_t ws_size,
                              hipStream_t stream) {
    const float* scale  = (const float*)d_in[0];   // [N0, 8] f32
    const int*   offset = (const int*)  d_in[1];   // [N0]    i32
    const int*   weight = (const int*)  d_in[2];   // [N0, 8] i32
    const float* act    = (const float*)d_in[3];   // [64, N1] f32
    float* out = (float*)d_out;

    const int N0 = in_sizes[1];        // rows (offset has one word per row)
    const int N1 = in_sizes[3] / MID;  // activation columns

    dim3 grid(N1 / BLOCK_N, N0 / BLOCK_M);
    w4a32_wmma_gemm<<<grid, dim3(256), 0, stream>>>(scale, offset, weight, act, out, N1);
}